// xLSTMRegressor_v2_71399536328859
// MI455X (gfx1250) — compile-verified
//
#include <hip/hip_runtime.h>
#include <math.h>

typedef float v2f __attribute__((ext_vector_type(2)));
typedef float v8f __attribute__((ext_vector_type(8)));

// ---------------- problem constants ----------------
constexpr int  B_    = 1024;
constexpr int  S_    = 20;
constexpr int  IN_   = 6;
constexpr int  E_    = 512;
constexpr int  NH_   = 4;
constexpr int  INNER_= 1024;     // 2*E
constexpr int  UP_   = 666;      // round(1.3*E)
constexpr int  UPP_  = 672;      // UP padded to mult of 32 (GEMM K-chunk)
constexpr int  UP2_  = 1332;     // 2*UP
constexpr long ROWS_ = (long)B_ * S_;   // 20480

// ---------------- workspace layout (floats) ----------------
constexpr long O_H    = 0;
constexpr long O_LNX  = O_H    + ROWS_ * E_;       // 10.49M
constexpr long O_XZ   = O_LNX  + ROWS_ * E_;       // +10.49M
constexpr long O_XC   = O_XZ   + ROWS_ * 2048L;    // +41.94M
constexpr long O_QKV  = O_XC   + ROWS_ * 1024L;    // +20.97M
constexpr long O_HATT = O_QKV  + ROWS_ * 3072L;    // +62.91M
constexpr long O_IGP  = O_HATT + ROWS_ * 1024L;    // +20.97M
constexpr long O_FGP  = O_IGP  + (long)B_ * NH_ * S_;
constexpr long O_HS   = O_FGP  + (long)B_ * NH_ * S_;
constexpr long O_CS   = O_HS   + (long)B_ * E_;
constexpr long O_NS   = O_CS   + (long)B_ * E_;
constexpr long O_MS   = O_NS   + (long)B_ * E_;
constexpr long O_REC  = O_MS   + (long)B_ * E_;    // 4 * B * E
constexpr long O_WP   = O_REC  + 4L * B_ * E_;     // 672*512

__device__ __forceinline__ float siluf(float x)  { return x / (1.f + expf(-x)); }
__device__ __forceinline__ float logsigf(float x){
    return (x >= 0.f) ? -log1pf(expf(-x)) : (x - log1pf(expf(x)));
}

// ======================================================================
// Generic batched strided GEMM: C = A @ W (+C), fp32 via V_WMMA_F32_16X16X4_F32.
// Block = 256 threads (8 waves), block tile 128(M) x 64(N).
// B tile (KC x 64) is staged in LDS once per block (register double-buffered),
// stored as (k,k+1) pairs so each wave fragment is a single ds_load_b64.
// Per-batch offsets: off = (bz/div)*s1 + (bz%div)*s2 for A, W, C.
// W element (k,n) at W[k*wks + n*wns]  (wks/wns swap => transposed weight).
// Requires: M % 128 == 0 (grid.x = M/128), K % 32 == 0. N guarded.
// ======================================================================
constexpr int KC_CH = 32;

__global__ __launch_bounds__(256) void k_gemm(
    const float* __restrict__ A, int adiv, long as1, long as2, int lda,
    const float* __restrict__ W, int wdiv, long wbs1, long wbs2, long wks, long wns,
    float*       __restrict__ C, int cdiv, long cs1, long cs2, int ldc,
    int N, int K, int addC)
{
    __shared__ float sB[2][KC_CH * 64];   // 16 KB

    const int bz = blockIdx.z;
    A += (long)(bz / adiv) * as1 + (long)(bz % adiv) * as2;
    W += (long)(bz / wdiv) * wbs1 + (long)(bz % wdiv) * wbs2;
    C += (long)(bz / cdiv) * cs1 + (long)(bz % cdiv) * cs2;

    const int wv   = threadIdx.x >> 5;
    const int lane = threadIdx.x & 31;
    const int l15  = lane & 15;
    const int hi   = lane >> 4;
    const long mrow = (long)blockIdx.x * 128 + wv * 16;
    const int  nbase = blockIdx.y * 64;
    const long r    = mrow + l15;
    const int  koff = hi * 2;              // ka = kk + koff is always even
    const int  c0   = nbase + l15;

    v8f acc0 = {}, acc1 = {}, acc2 = {}, acc3 = {};
    float rg[8];

    // cooperative staging: 2048 elements (KC_CH x 64) -> 8 per thread
    auto load_regs = [&](int kc0) {
#pragma unroll
        for (int i = 0; i < 8; ++i) {
            int e = (int)threadIdx.x + i * 256;
            int k = e >> 6, c = e & 63;
            int  gc  = nbase + c;
            bool ok  = gc < N;
            int  gcc = ok ? gc : 0;
            float v  = W[(long)(kc0 + k) * wks + (long)gcc * wns];
            rg[i] = ok ? v : 0.f;
        }
    };
    auto store_lds = [&](int buf) {
#pragma unroll
        for (int i = 0; i < 8; ++i) {
            int e = (int)threadIdx.x + i * 256;
            int k = e >> 6, c = e & 63;
            sB[buf][(k >> 1) * 128 + c * 2 + (k & 1)] = rg[i];
        }
    };

    load_regs(0);
    store_lds(0);
    __syncthreads();

    const int nch = K / KC_CH;
    for (int ci = 0; ci < nch; ++ci) {
        const int cur = ci & 1;
        if (ci + 1 < nch) load_regs((ci + 1) * KC_CH);
        if ((ci + 2) * KC_CH < K)
            __builtin_prefetch(&A[r * (long)lda + (ci + 2) * KC_CH], 0, 3);

        const int kc0 = ci * KC_CH;
#pragma unroll
        for (int kk = 0; kk < KC_CH; kk += 4) {
            const int ka = kk + koff;      // even
            v2f a;
            a.x = A[r * (long)lda + kc0 + ka];
            a.y = A[r * (long)lda + kc0 + ka + 1];
            const float* bp = &sB[cur][(ka >> 1) * 128];
            v2f b0 = *(const v2f*)(bp + (l15     ) * 2);
            v2f b1 = *(const v2f*)(bp + (l15 + 16) * 2);
            v2f b2 = *(const v2f*)(bp + (l15 + 32) * 2);
            v2f b3 = *(const v2f*)(bp + (l15 + 48) * 2);
            acc0 = __builtin_amdgcn_wmma_f32_16x16x4_f32(false, a, false, b0, (short)0, acc0, false, false);
            acc1 = __builtin_amdgcn_wmma_f32_16x16x4_f32(false, a, false, b1, (short)0, acc1, false, false);
            acc2 = __builtin_amdgcn_wmma_f32_16x16x4_f32(false, a, false, b2, (short)0, acc2, false, false);
            acc3 = __builtin_amdgcn_wmma_f32_16x16x4_f32(false, a, false, b3, (short)0, acc3, false, false);
        }
        if (ci + 1 < nch) {
            __syncthreads();
            store_lds((ci + 1) & 1);
            __syncthreads();
        }
    }

    const long rb = mrow + hi * 8;
#pragma unroll
    for (int v = 0; v < 8; ++v) {
        const long rr = rb + v;
        int c;
        c = c0;      if (c < N) { long o = rr * (long)ldc + c; C[o] = addC ? C[o] + acc0[v] : acc0[v]; }
        c = c0 + 16; if (c < N) { long o = rr * (long)ldc + c; C[o] = addC ? C[o] + acc1[v] : acc1[v]; }
        c = c0 + 32; if (c < N) { long o = rr * (long)ldc + c; C[o] = addC ? C[o] + acc2[v] : acc2[v]; }
        c = c0 + 48; if (c < N) { long o = rr * (long)ldc + c; C[o] = addC ? C[o] + acc3[v] : acc3[v]; }
    }
}

// ======================================================================
// Embedding: H = x @ W_emb + b_emb + pos
// ======================================================================
__global__ void k_embed(const float* __restrict__ x, const float* __restrict__ We,
                        const float* __restrict__ be, const float* __restrict__ pos,
                        float* __restrict__ H)
{
    long i = (long)blockIdx.x * 256 + threadIdx.x;
    if (i >= ROWS_ * E_) return;
    int  e  = (int)(i % E_);
    long bs = i / E_;
    int  s  = (int)(bs % S_);
    const float* xr = x + bs * IN_;
    float a = be[e] + pos[(long)s * E_ + e];
#pragma unroll
    for (int k = 0; k < IN_; ++k) a += xr[k] * We[(long)k * E_ + e];
    H[i] = a;
}

// ======================================================================
// LayerNorm over C per row, with weight
// ======================================================================
__global__ __launch_bounds__(256) void k_ln(const float* __restrict__ x,
                                            const float* __restrict__ w,
                                            float* __restrict__ o, int C)
{
    long r = blockIdx.x;
    const float* xr = x + r * (long)C;
    float s = 0.f, s2 = 0.f;
    for (int c = threadIdx.x; c < C; c += 256) { float v = xr[c]; s += v; s2 += v * v; }
    __shared__ float sm[256], sm2[256];
    sm[threadIdx.x] = s; sm2[threadIdx.x] = s2; __syncthreads();
    for (int off = 128; off > 0; off >>= 1) {
        if (threadIdx.x < off) { sm[threadIdx.x] += sm[threadIdx.x + off]; sm2[threadIdx.x] += sm2[threadIdx.x + off]; }
        __syncthreads();
    }
    float mu  = sm[0] / C;
    float var = sm2[0] / C - mu * mu;
    float rs  = rsqrtf(var + 1e-5f);
    for (int c = threadIdx.x; c < C; c += 256) o[r * (long)C + c] = (xr[c] - mu) * rs * w[c];
}

// ======================================================================
// Causal depthwise conv (K=4) + SiLU. Input row stride ldx (xi lives inside XZ).
// ======================================================================
__global__ void k_conv_silu(const float* __restrict__ x, const float* __restrict__ w,
                            const float* __restrict__ bc, float* __restrict__ y,
                            int C, int ldx, long total)
{
    long i = (long)blockIdx.x * 256 + threadIdx.x;
    if (i >= total) return;
    int  c  = (int)(i % C);
    long bs = i / C;
    int  s  = (int)(bs % S_);
    long bb = bs / S_;
    float acc = bc[c];
#pragma unroll
    for (int k = 0; k < 4; ++k) {
        int sp = s - 3 + k;
        if (sp >= 0) acc += w[c * 4 + k] * x[(bb * S_ + sp) * (long)ldx + c];
    }
    y[bs * (long)C + c] = siluf(acc);
}

// ======================================================================
// mLSTM head-wise 4x4 blocks: out[r, n*4+o] = sum_i x[r, n*4+i] * w[n][o][i]
// ======================================================================
__global__ void k_headwise4(const float* __restrict__ x, int ldx,
                            const float* __restrict__ w,
                            float* __restrict__ out, int ldo, int coloff)
{
    long i = (long)blockIdx.x * 256 + threadIdx.x;
    if (i >= ROWS_ * 1024L) return;
    int  c = (int)(i % 1024);
    long r = i / 1024;
    int  n = c >> 2, oo = c & 3;
    const float* xr = x + r * (long)ldx + n * 4;
    const float* wr = w + n * 16 + oo * 4;
    out[r * (long)ldo + coloff + c] = xr[0]*wr[0] + xr[1]*wr[1] + xr[2]*wr[2] + xr[3]*wr[3];
}

// ======================================================================
// Gate projections: igp/fgp = qkv @ W(3072x4) + b, transposed to (b,h,s)
// ======================================================================
__global__ __launch_bounds__(256) void k_gates(const float* __restrict__ qkv,
    const float* __restrict__ Wig, const float* __restrict__ big,
    const float* __restrict__ Wfg, const float* __restrict__ bfg,
    float* __restrict__ igp, float* __restrict__ fgp)
{
    long r = blockIdx.x;           // b*S + s
    int  s = (int)(r % S_);
    long b = r / S_;
    const float* xr = qkv + r * 3072L;
    float ai[4] = {0,0,0,0}, af[4] = {0,0,0,0};
    for (int k = threadIdx.x; k < 3072; k += 256) {
        float v = xr[k];
#pragma unroll
        for (int h = 0; h < 4; ++h) { ai[h] += v * Wig[k * 4 + h]; af[h] += v * Wfg[k * 4 + h]; }
    }
    __shared__ float smi[4][256], smf[4][256];
#pragma unroll
    for (int h = 0; h < 4; ++h) { smi[h][threadIdx.x] = ai[h]; smf[h][threadIdx.x] = af[h]; }
    __syncthreads();
    for (int off = 128; off > 0; off >>= 1) {
        if (threadIdx.x < off)
#pragma unroll
            for (int h = 0; h < 4; ++h) {
                smi[h][threadIdx.x] += smi[h][threadIdx.x + off];
                smf[h][threadIdx.x] += smf[h][threadIdx.x + off];
            }
        __syncthreads();
    }
    if (threadIdx.x < 4) {
        int h = threadIdx.x;
        igp[(b * 4 + h) * S_ + s] = smi[h][0] + big[h];
        fgp[(b * 4 + h) * S_ + s] = smf[h][0] + bfg[h];
    }
}

// ======================================================================
// Fused mLSTM attention per (b,h): stabilized log-domain D matrix.
// ======================================================================
__global__ __launch_bounds__(256) void k_mlstm_att(const float* __restrict__ qkv,
    const float* __restrict__ igp, const float* __restrict__ fgp,
    float* __restrict__ hatt)
{
    __shared__ float sq[S_ * 256];
    __shared__ float sk[S_ * 256];
    __shared__ float sv[S_ * 256];
    __shared__ float cum[S_], igs[S_];
    __shared__ float qkr[32], coef[32];

    int  bh = blockIdx.x;          // b*NH + h
    int  h  = bh & 3;
    long b  = bh >> 2;
    int  tid = threadIdx.x;

    for (int i = tid; i < S_ * 256; i += 256) {
        int s = i >> 8, d = i & 255;
        long base = (b * S_ + s) * 3072L + (long)h * 256 + d;
        sq[i] = qkv[base];
        sk[i] = qkv[base + 1024];
        sv[i] = qkv[base + 2048];
    }
    if (tid == 0) {
        float run = 0.f;
        for (int s = 0; s < S_; ++s) {
            run += logsigf(fgp[(long)bh * S_ + s]);
            cum[s] = run;
            igs[s] = igp[(long)bh * S_ + s];
        }
    }
    __syncthreads();

    const int t8 = tid >> 3;   // 0..31 (t index)
    const int j8 = tid & 7;
    for (int s = 0; s < S_; ++s) {
        if (t8 <= s) {
            float p = 0.f;
            const float* qrow = sq + s * 256 + j8 * 32;
            const float* krow = sk + t8 * 256 + j8 * 32;
#pragma unroll
            for (int d = 0; d < 32; ++d) p += qrow[d] * krow[d];
            p += __shfl_xor(p, 1, 32);
            p += __shfl_xor(p, 2, 32);
            p += __shfl_xor(p, 4, 32);
            if (j8 == 0) qkr[t8] = p * 0.0625f;  // / sqrt(DHM)=16
        }
        __syncthreads();
        if (tid == 0) {
            float maxD = -3.0e38f;
            for (int t = 0; t <= s; ++t) {
                float ld = cum[s] - cum[t] + igs[t];
                if (ld > maxD) maxD = ld;
            }
            float csum = 0.f;
            for (int t = 0; t <= s; ++t) {
                float cm = qkr[t] * expf(cum[s] - cum[t] + igs[t] - maxD);
                coef[t] = cm;
                csum += cm;
            }
            float norm = fmaxf(fabsf(csum), expf(-maxD)) + 1e-6f;
            float inv = 1.f / norm;
            for (int t = 0; t <= s; ++t) coef[t] *= inv;
        }
        __syncthreads();
        {
            float a = 0.f;
            for (int t = 0; t <= s; ++t) a += coef[t] * sv[t * 256 + tid];
            hatt[(b * S_ + s) * (long)INNER_ + (long)h * 256 + tid] = a;
        }
        __syncthreads();
    }
}

// ======================================================================
// mLSTM groupnorm (groups of 256) + skip + SiLU(z) gating
// ======================================================================
__global__ __launch_bounds__(256) void k_gn_gate(const float* __restrict__ hatt,
    const float* __restrict__ xc, const float* __restrict__ xz,
    const float* __restrict__ nw, const float* __restrict__ skip,
    float* __restrict__ out)
{
    long rg = blockIdx.x;          // row*4 + g
    long r  = rg >> 2;
    int  g  = (int)(rg & 3);
    int  c  = g * 256 + threadIdx.x;
    float v = hatt[r * 1024L + c];
    __shared__ float sm[256], sm2[256];
    sm[threadIdx.x] = v; sm2[threadIdx.x] = v * v; __syncthreads();
    for (int off = 128; off > 0; off >>= 1) {
        if (threadIdx.x < off) { sm[threadIdx.x] += sm[threadIdx.x + off]; sm2[threadIdx.x] += sm2[threadIdx.x + off]; }
        __syncthreads();
    }
    float mu  = sm[0] / 256.f;
    float var = sm2[0] / 256.f - mu * mu;
    float xn  = (v - mu) * rsqrtf(var + 1e-5f);
    float z   = xz[r * 2048L + 1024 + c];
    out[r * 1024L + c] = (xn * nw[c] + skip[c] * xc[r * 1024L + c]) * siluf(z);
}

// ======================================================================
// sLSTM state init / per-step pointwise update
// ======================================================================
__global__ void k_slstm_init(float* hS, float* cS, float* nS, float* mS)
{
    long i = (long)blockIdx.x * 256 + threadIdx.x;
    if (i >= (long)B_ * E_) return;
    hS[i] = 0.f; cS[i] = 0.f; nS[i] = 0.f; mS[i] = -1e30f;
}

__global__ void k_slstm_step(const float* __restrict__ Gi, const float* __restrict__ Gf,
    const float* __restrict__ Gz, const float* __restrict__ Go,
    const float* __restrict__ rec, const float* __restrict__ bias,
    float* __restrict__ hS, float* __restrict__ cS,
    float* __restrict__ nS, float* __restrict__ mS,
    float* __restrict__ hs_out, int t)
{
    long i = (long)blockIdx.x * 256 + threadIdx.x;
    if (i >= (long)B_ * E_) return;
    long b   = i >> 9;
    int  idx = (int)(i & 511);
    int  hh  = idx >> 7, d = idx & 127;
    long gidx = (b * S_ + t) * (long)E_ + idx;
    const long BE = (long)B_ * E_;
    float ir  = Gi[gidx] + rec[0 * BE + i] + bias[(0 * 4 + hh) * 128 + d];
    float fr  = Gf[gidx] + rec[1 * BE + i] + bias[(1 * 4 + hh) * 128 + d];
    float zr  = Gz[gidx] + rec[2 * BE + i] + bias[(2 * 4 + hh) * 128 + d];
    float orr = Go[gidx] + rec[3 * BE + i] + bias[(3 * 4 + hh) * 128 + d];
    float m    = mS[i];
    float lfm  = m + logsigf(fr);
    float mnew = fmaxf(ir, lfm);
    float ig   = expf(ir - mnew);
    float fg   = expf(lfm - mnew);
    float cn   = fg * cS[i] + ig * tanhf(zr);
    float nn   = fg * nS[i] + ig;
    float hn   = (1.f / (1.f + expf(-orr))) * cn / nn;
    cS[i] = cn; nS[i] = nn; mS[i] = mnew; hS[i] = hn;
    hs_out[gidx] = hn;
}

// ======================================================================
// sLSTM groupnorm (groups of 128) + residual add into H
// ======================================================================
__global__ __launch_bounds__(128) void k_gn_res(const float* __restrict__ x,
    const float* __restrict__ w, float* __restrict__ H)
{
    long rg = blockIdx.x;          // row*4 + g
    long r  = rg >> 2;
    int  g  = (int)(rg & 3);
    int  c  = g * 128 + threadIdx.x;
    float v = x[r * (long)E_ + c];
    __shared__ float sm[128], sm2[128];
    sm[threadIdx.x] = v; sm2[threadIdx.x] = v * v; __syncthreads();
    for (int off = 64; off > 0; off >>= 1) {
        if (threadIdx.x < off) { sm[threadIdx.x] += sm[threadIdx.x + off]; sm2[threadIdx.x] += sm2[threadIdx.x + off]; }
        __syncthreads();
    }
    float mu  = sm[0] / 128.f;
    float var = sm2[0] / 128.f - mu * mu;
    H[r * (long)E_ + c] += (v - mu) * rsqrtf(var + 1e-5f) * w[c];
}

// ======================================================================
// FFN: exact-GELU(gate) * up, written into K-padded buffer (672 cols)
// ======================================================================
__global__ void k_geglu(const float* __restrict__ gu, float* __restrict__ gact)
{
    long i = (long)blockIdx.x * 256 + threadIdx.x;
    if (i >= ROWS_ * (long)UPP_) return;
    int  c = (int)(i % UPP_);
    long r = i / UPP_;
    float o = 0.f;
    if (c < UP_) {
        float g = gu[r * (long)UP2_ + c];
        float u = gu[r * (long)UP2_ + UP_ + c];
        o = 0.5f * g * (1.f + erff(g * 0.70710678118654752f)) * u;
    }
    gact[i] = o;
}

__global__ void k_padW(const float* __restrict__ Wd, float* __restrict__ Wp)
{
    long i = (long)blockIdx.x * 256 + threadIdx.x;
    if (i >= (long)UPP_ * E_) return;
    long k = i / E_;
    Wp[i] = (k < UP_) ? Wd[i] : 0.f;
}

// ======================================================================
// Final head: last timestep -> @W1 + b1 -> @W2 + b2
// ======================================================================
__global__ __launch_bounds__(64) void k_head(const float* __restrict__ lnp,
    const float* __restrict__ W1, const float* __restrict__ b1,
    const float* __restrict__ W2, const float* __restrict__ b2,
    float* __restrict__ out)
{
    long b = blockIdx.x;
    const float* xr = lnp + (b * S_ + (S_ - 1)) * (long)E_;
    __shared__ float y1[48];
    int j = threadIdx.x;
    if (j < 48) {
        float a = b1[j];
        for (int k = 0; k < E_; ++k) a += xr[k] * W1[(long)k * 48 + j];
        y1[j] = a;
    }
    __syncthreads();
    if (j == 0) {
        float a = b2[0];
        for (int t = 0; t < 48; ++t) a += y1[t] * W2[t];
        out[b] = a;
    }
}

// ======================================================================
// Host orchestration
// ======================================================================
static inline void gemm(hipStream_t st,
    const float* A, int adiv, long as1, long as2, int lda,
    const float* W, int wdiv, long wbs1, long wbs2, long wks, long wns,
    float* C, int cdiv, long cs1, long cs2, int ldc,
    int M, int N, int K, int addC, int batch)
{
    dim3 g((unsigned)(M / 128), (unsigned)((N + 63) / 64), (unsigned)batch);
    k_gemm<<<g, dim3(256), 0, st>>>(A, adiv, as1, as2, lda,
                                    W, wdiv, wbs1, wbs2, wks, wns,
                                    C, cdiv, cs1, cs2, ldc, N, K, addC);
}

static inline dim3 g1(long n) { return dim3((unsigned)((n + 255) / 256)); }

extern "C" void kernel_launch(void* const* d_in, const int* in_sizes, int n_in,
                              void* d_out, int out_size, void* d_ws, size_t ws_size,
                              hipStream_t stream)
{
    (void)in_sizes; (void)n_in; (void)out_size; (void)ws_size;
    const float* x       = (const float*)d_in[0];
    const float* W_emb   = (const float*)d_in[1];
    const float* b_emb   = (const float*)d_in[2];
    const float* pos     = (const float*)d_in[3];
    const float* ln0     = (const float*)d_in[4];
    const float* ln1     = (const float*)d_in[5];
    const float* ln_ff   = (const float*)d_in[6];
    const float* ln_post = (const float*)d_in[7];
    const float* mWup    = (const float*)d_in[8];
    const float* mconvw  = (const float*)d_in[9];
    const float* mconvb  = (const float*)d_in[10];
    const float* mWq     = (const float*)d_in[11];
    const float* mWk     = (const float*)d_in[12];
    const float* mWv     = (const float*)d_in[13];
    const float* mWig    = (const float*)d_in[14];
    const float* mbig    = (const float*)d_in[15];
    const float* mWfg    = (const float*)d_in[16];
    const float* mbfg    = (const float*)d_in[17];
    const float* mnormw  = (const float*)d_in[18];
    const float* mskip   = (const float*)d_in[19];
    const float* mWdown  = (const float*)d_in[20];
    const float* sconvw  = (const float*)d_in[21];
    const float* sconvb  = (const float*)d_in[22];
    const float* sWi     = (const float*)d_in[23];
    const float* sWf     = (const float*)d_in[24];
    const float* sWz     = (const float*)d_in[25];
    const float* sWo     = (const float*)d_in[26];
    const float* sR      = (const float*)d_in[27];
    const float* sbias   = (const float*)d_in[28];
    const float* snormw  = (const float*)d_in[29];
    const float* fWup    = (const float*)d_in[30];
    const float* fWdown  = (const float*)d_in[31];
    const float* W1      = (const float*)d_in[32];
    const float* b1      = (const float*)d_in[33];
    const float* W2      = (const float*)d_in[34];
    const float* b2      = (const float*)d_in[35];

    float* ws   = (float*)d_ws;
    float* H    = ws + O_H;
    float* LNX  = ws + O_LNX;
    float* XZ   = ws + O_XZ;
    float* XC   = ws + O_XC;
    float* QKV  = ws + O_QKV;
    float* HATT = ws + O_HATT;
    float* IGP  = ws + O_IGP;
    float* FGP  = ws + O_FGP;
    float* hS   = ws + O_HS;
    float* cS   = ws + O_CS;
    float* nS   = ws + O_NS;
    float* mS   = ws + O_MS;
    float* REC  = ws + O_REC;
    float* WP   = ws + O_WP;
    float* out  = (float*)d_out;

    // ---- embedding + positional ----
    k_embed<<<g1(ROWS_ * E_), 256, 0, stream>>>(x, W_emb, b_emb, pos, H);

    // ================= mLSTM layer =================
    k_ln<<<dim3((unsigned)ROWS_), 256, 0, stream>>>(H, ln0, LNX, E_);
    // xz = ln0 @ Wup  (20480 x 512 -> 2048)
    gemm(stream, LNX, 1, 0, 0, E_, mWup, 1, 0, 0, 2048, 1,
         XZ, 1, 0, 0, 2048, (int)ROWS_, 2048, E_, 0, 1);
    // xc = silu(conv(xi))   (xi = XZ[:, :1024])
    k_conv_silu<<<g1(ROWS_ * 1024L), 256, 0, stream>>>(XZ, mconvw, mconvb, XC, 1024, 2048, ROWS_ * 1024L);
    // q/k/v headwise 4x4 blocks into QKV concat
    k_headwise4<<<g1(ROWS_ * 1024L), 256, 0, stream>>>(XC, 1024, mWq, QKV, 3072, 0);
    k_headwise4<<<g1(ROWS_ * 1024L), 256, 0, stream>>>(XC, 1024, mWk, QKV, 3072, 1024);
    k_headwise4<<<g1(ROWS_ * 1024L), 256, 0, stream>>>(XZ, 2048, mWv, QKV, 3072, 2048);
    // gate projections
    k_gates<<<dim3((unsigned)ROWS_), 256, 0, stream>>>(QKV, mWig, mbig, mWfg, mbfg, IGP, FGP);
    // fused attention per (b, h)
    k_mlstm_att<<<dim3((unsigned)(B_ * NH_)), 256, 0, stream>>>(QKV, IGP, FGP, HATT);
    // groupnorm + skip + silu(z) gating -> reuse QKV region as GATED
    k_gn_gate<<<dim3((unsigned)(ROWS_ * 4)), 256, 0, stream>>>(HATT, XC, XZ, mnormw, mskip, QKV);
    // H += GATED @ Wdown (1024 -> 512)
    gemm(stream, QKV, 1, 0, 0, 1024, mWdown, 1, 0, 0, E_, 1,
         H, 1, 0, 0, E_, (int)ROWS_, E_, 1024, 1, 1);

    // ================= sLSTM layer =================
    k_ln<<<dim3((unsigned)ROWS_), 256, 0, stream>>>(H, ln1, LNX, E_);
    k_conv_silu<<<g1(ROWS_ * (long)E_), 256, 0, stream>>>(LNX, sconvw, sconvb, XC, E_, E_, ROWS_ * (long)E_);
    // gate pre-activations, headwise 128x128 (transposed weights 'noi'), batch over heads
    float* Gi = XZ;
    float* Gf = XZ + ROWS_ * (long)E_;
    float* Gz = XZ + 2 * ROWS_ * (long)E_;
    float* Go = XZ + 3 * ROWS_ * (long)E_;
    gemm(stream, XC,  1, 128, 0, E_, sWi, 1, 16384, 0, 1, 128, Gi, 1, 128, 0, E_, (int)ROWS_, 128, 128, 0, 4);
    gemm(stream, XC,  1, 128, 0, E_, sWf, 1, 16384, 0, 1, 128, Gf, 1, 128, 0, E_, (int)ROWS_, 128, 128, 0, 4);
    gemm(stream, LNX, 1, 128, 0, E_, sWz, 1, 16384, 0, 1, 128, Gz, 1, 128, 0, E_, (int)ROWS_, 128, 128, 0, 4);
    gemm(stream, LNX, 1, 128, 0, E_, sWo, 1, 16384, 0, 1, 128, Go, 1, 128, 0, E_, (int)ROWS_, 128, 128, 0, 4);
    // scan
    float* HSOUT = QKV;   // (B,S,E) reuse
    k_slstm_init<<<g1((long)B_ * E_), 256, 0, stream>>>(hS, cS, nS, mS);
    for (int t = 0; t < S_; ++t) {
        // rec[g,b,h,:] = h_state[b,h,:] @ R[g,h]   batch bz = g*4+h
        gemm(stream, hS, 4, 0, 128, E_,
             sR, 1, 16384, 0, 128, 1,
             REC, 4, (long)B_ * E_, 128, E_,
             B_, 128, 128, 0, 16);
        k_slstm_step<<<g1((long)B_ * E_), 256, 0, stream>>>(Gi, Gf, Gz, Go, REC, sbias,
                                                            hS, cS, nS, mS, HSOUT, t);
    }
    // groupnorm + residual
    k_gn_res<<<dim3((unsigned)(ROWS_ * 4)), 128, 0, stream>>>(HSOUT, snormw, H);

    // ================= FFN =================
    k_ln<<<dim3((unsigned)ROWS_), 256, 0, stream>>>(H, ln_ff, LNX, E_);
    float* GU   = XZ;   // (ROWS, 1332)
    float* GACT = XC;   // (ROWS, 672) padded
    gemm(stream, LNX, 1, 0, 0, E_, fWup, 1, 0, 0, UP2_, 1,
         GU, 1, 0, 0, UP2_, (int)ROWS_, UP2_, E_, 0, 1);
    k_geglu<<<g1(ROWS_ * (long)UPP_), 256, 0, stream>>>(GU, GACT);
    k_padW<<<g1((long)UPP_ * E_), 256, 0, stream>>>(fWdown, WP);
    gemm(stream, GACT, 1, 0, 0, UPP_, WP, 1, 0, 0, E_, 1,
         H, 1, 0, 0, E_, (int)ROWS_, E_, UPP_, 1, 1);

    // ================= head =================
    k_ln<<<dim3((unsigned)ROWS_), 256, 0, stream>>>(H, ln_post, LNX, E_);
    k_head<<<dim3((unsigned)B_), 64, 0, stream>>>(LNX, W1, b1, W2, b2, out);
}